// RandomTransformerModel_11716670783706
// MI455X (gfx1250) — compile-verified
//
#include <hip/hip_runtime.h>
#include <hip/hip_bf16.h>

typedef __attribute__((ext_vector_type(2))) float v2f;
typedef __attribute__((ext_vector_type(8))) float v8f;

constexpr int B_ = 4, L_ = 1024, VOCAB_ = 50257, D_ = 64;

// Kernel 1: per-batch softmax of last attention row, weighted embedding
// gather-sum m[d] = sum_l w[l]*E_w[d, x[l]], then c = m @ V_w.T  -> ws
__global__ __launch_bounds__(256) void pool_embed_kernel(
    const int* __restrict__ x, const float* __restrict__ a,
    const float* __restrict__ E_w, const float* __restrict__ V_w,
    float* __restrict__ c_ws)
{
    __shared__ float w[L_];
    __shared__ int   xs[L_];
    __shared__ float red[256];
    __shared__ float m_s[D_];

    const int b = blockIdx.x;
    const int t = threadIdx.x;
    const float* arow = a + ((size_t)b * L_ + (L_ - 1)) * (size_t)L_;

    // load the 1024-wide row, cache token ids, local max
    float vals[4];
    float lmax = -INFINITY;
    #pragma unroll
    for (int i = 0; i < 4; ++i) {
        int l = t + i * 256;
        float v = arow[l];
        vals[i] = v;
        lmax = fmaxf(lmax, v);
        xs[l] = x[b * L_ + l];
    }
    red[t] = lmax;
    __syncthreads();
    for (int s = 128; s > 0; s >>= 1) {
        if (t < s) red[t] = fmaxf(red[t], red[t + s]);
        __syncthreads();
    }
    float mx = red[0];
    __syncthreads();

    float lsum = 0.f;
    #pragma unroll
    for (int i = 0; i < 4; ++i) {
        int l = t + i * 256;
        float e = __expf(vals[i] - mx);
        w[l] = e;                 // unnormalized; fold 1/sum in later
        lsum += e;
    }
    red[t] = lsum;
    __syncthreads();
    for (int s = 128; s > 0; s >>= 1) {
        if (t < s) red[t] += red[t + s];
        __syncthreads();
    }
    float inv_sum = 1.0f / red[0];
    __syncthreads();

    // m[d] = sum_l w[l] * E_w[d*VOCAB + xs[l]]  (64 dims x 4 token-chunks)
    int d = t & 63;
    int chunk = t >> 6;
    const float* Ed = E_w + (size_t)d * VOCAB_;
    float partial = 0.f;
    int l0 = chunk * 256;
    for (int l = l0; l < l0 + 256; ++l)
        partial += w[l] * Ed[xs[l]];
    red[t] = partial;
    __syncthreads();
    if (t < 64)
        m_s[t] = (red[t] + red[t + 64] + red[t + 128] + red[t + 192]) * inv_sum;
    __syncthreads();

    // c[b][e] = sum_d m[d] * V_w[e*64 + d]
    if (t < 64) {
        const float* Vr = V_w + t * D_;
        float cc = 0.f;
        #pragma unroll
        for (int dd = 0; dd < D_; ++dd) cc += m_s[dd] * Vr[dd];
        c_ws[b * D_ + t] = cc;
    }
}

// Kernel 2: out[0..3, :] = c @ U_w.T via V_WMMA_F32_16X16X4_F32.
// One wave per 16-column vocab tile; K=64 as 16 steps of K=4.
// All loads are unconditional (addresses clamped/wrapped): garbage in
// B-column n or A-row m only corrupts the matching D entries, which are
// never stored.
__global__ __launch_bounds__(256) void vocab_wmma_kernel(
    const float* __restrict__ U_w, const float* __restrict__ c_ws,
    float* __restrict__ out)
{
    const int t    = threadIdx.x;
    const int wave = t >> 5;
    const int lane = t & 31;
    const int tile = blockIdx.x * 8 + wave;
    const int col0 = tile * 16;
    if (col0 >= VOCAB_) return;          // wave-uniform exit

    const int n     = lane & 15;         // B/D column within tile
    const int khalf = lane >> 4;         // lanes 16-31 handle K+2,K+3
    const int col   = col0 + n;
    const bool cvalid = col < VOCAB_;
    const int colc  = cvalid ? col : (VOCAB_ - 1);   // clamped, in-bounds
    const int mrow  = lane & 15;                     // A-matrix row (M)

    const float* Urow = U_w  + (size_t)colc * D_;    // B: U_w.T column = U_w row
    const float* Crow = c_ws + (mrow & 3) * D_;      // A: rows 4..15 wrap (unused)

    // Each lane pulls its half of the K=64 strip as 16 aligned float2 loads.
    v2f afr[16], bfr[16];
    #pragma unroll
    for (int kk = 0; kk < 16; ++kk) {
        const int kb = kk * 4 + 2 * khalf;
        afr[kk] = *(const v2f*)(Crow + kb);
        bfr[kk] = *(const v2f*)(Urow + kb);
    }

    v8f acc = {};
    #pragma unroll
    for (int kk = 0; kk < 16; ++kk) {
        acc = __builtin_amdgcn_wmma_f32_16x16x4_f32(
            false, afr[kk], false, bfr[kk], (short)0, acc, false, false);
    }

    // C/D layout: VGPR r, lanes 0-15 hold row M=r -> batches 0..3
    if (khalf == 0 && cvalid) {
        #pragma unroll
        for (int r = 0; r < B_; ++r)
            out[(size_t)r * VOCAB_ + col] = acc[r];
    }
}

extern "C" void kernel_launch(void* const* d_in, const int* in_sizes, int n_in,
                              void* d_out, int out_size, void* d_ws, size_t ws_size,
                              hipStream_t stream) {
    const int*   x   = (const int*)d_in[0];
    const float* a   = (const float*)d_in[1];
    const float* E_w = (const float*)d_in[2];
    const float* V_w = (const float*)d_in[3];
    const float* U_w = (const float*)d_in[4];
    float* out  = (float*)d_out;
    float* c_ws = (float*)d_ws;   // 256 floats

    pool_embed_kernel<<<B_, 256, 0, stream>>>(x, a, E_w, V_w, c_ws);

    const int ntiles  = (VOCAB_ + 15) / 16;   // 3142
    const int nblocks = (ntiles + 7) / 8;     // 393
    vocab_wmma_kernel<<<nblocks, 256, 0, stream>>>(U_w, c_ws, out);
}